// SoftGVQLayer_28046136443278
// MI455X (gfx1250) — compile-verified
//
#include <hip/hip_runtime.h>
#include <hip/hip_bf16.h>

typedef __attribute__((ext_vector_type(2))) float v2f;
typedef __attribute__((ext_vector_type(8))) float v8f;

#define NB     32
#define NL     2048
#define EMB    512
#define VQ     32
#define MCODES 32
#define LOG2N  16
#define NTOK   (NB * NL)
#define WAVES  8
#define TPB    256
#define TOK_PER_BLOCK (WAVES * 16)
#define SCR_STRIDE 34   // floats; even -> 8B-aligned b64 frags; odd*17 banks -> no conflicts

__global__ __launch_bounds__(TPB)
void softgvq_fused(const float* __restrict__ h,
                   const int*   __restrict__ attn_mask,
                   const float* __restrict__ Wp,   // [512][32]
                   const float* __restrict__ bp,   // [32]
                   const float* __restrict__ Wi,   // [32][512]
                   const float* __restrict__ bi,   // [512]
                   const float* __restrict__ emb,  // [32][32]
                   float* __restrict__ outQ,
                   long long* __restrict__ outCode,
                   float* __restrict__ outLoss)
{
    __shared__ __align__(16) float POOL[8192];              // swizzled W fragments (reused)
    __shared__ __align__(16) float EL[MCODES * VQ];         // normalized codebook
    __shared__ __align__(16) float E2[MCODES];              // row norms^2 after normalize
    __shared__ __align__(16) float SCR[WAVES * 16 * SCR_STRIDE]; // per-wave 16x32 scratch

    const int tid  = threadIdx.x;
    const int wave = tid >> 5;
    const int lane = tid & 31;
    const int col  = lane & 15;   // N-col / token row selector
    const int kh   = lane >> 4;   // K half
    const int tokBase = blockIdx.x * TOK_PER_BLOCK + wave * 16;

    // ---- stage + normalize codebook (32x32, tiny) ----
    for (int f = tid; f < MCODES * VQ; f += TPB) EL[f] = emb[f];
    __syncthreads();
    if (tid < MCODES) {
        float s = 0.f;
        for (int d = 0; d < VQ; ++d) { float v = EL[tid * VQ + d]; s += v * v; }
        float inv = 1.0f / (sqrtf(s) + 1e-8f);
        float s2 = 0.f;
        for (int d = 0; d < VQ; ++d) {
            float v = EL[tid * VQ + d] * inv; EL[tid * VQ + d] = v; s2 += v * v;
        }
        E2[tid] = s2;
    }

    // ---- GEMM1: hp[16x32] = h[16x512] @ Wp[512x32]  (v_wmma_f32_16x16x4_f32) ----
    v8f acc0 = {}; v8f acc1 = {};
    const float* hrow = h + (size_t)(tokBase + col) * EMB + 2 * kh;
    for (int half = 0; half < 2; ++half) {
        __syncthreads();
        // swizzle-stage 64 k-blocks of Wp into B-fragment order (32 KB)
        for (int f = tid; f < 8192; f += TPB) {
            int j = f & 1, ln = (f >> 1) & 31, tl = (f >> 6) & 1, kb = f >> 7;
            int kg = 4 * (half * 64 + kb) + 2 * (ln >> 4) + j;
            POOL[f] = Wp[kg * VQ + tl * 16 + (ln & 15)];
        }
        __syncthreads();
        const float* hptr = hrow + half * 256;
        #pragma unroll 4
        for (int kb = 0; kb < 64; ++kb) {
            v2f a  = *(const v2f*)(hptr + 4 * kb);                  // global_load_b64
            v2f b0 = *(const v2f*)&POOL[kb * 128 + lane * 2];       // ds_load_b64
            v2f b1 = *(const v2f*)&POOL[kb * 128 + 64 + lane * 2];
            acc0 = __builtin_amdgcn_wmma_f32_16x16x4_f32(false, a, false, b0, (short)0, acc0, false, false);
            acc1 = __builtin_amdgcn_wmma_f32_16x16x4_f32(false, a, false, b1, (short)0, acc1, false, false);
        }
    }

    // ---- dump hp (+b_proj) to per-wave LDS scratch in C/D layout ----
    float bp0 = bp[col], bp1 = bp[16 + col];
    float* scrw = &SCR[wave * 16 * SCR_STRIDE];
    #pragma unroll
    for (int r = 0; r < 8; ++r) {
        int tok = r + 8 * kh;
        scrw[tok * SCR_STRIDE + col]      = acc0[r] + bp0;
        scrw[tok * SCR_STRIDE + 16 + col] = acc1[r] + bp1;
    }
    __syncthreads();

    // ---- scalar phase: one token per lane (lanes 0..15) ----
    if (lane < 16) {
        float* row = &scrw[col * SCR_STRIDE];
        float hp[VQ]; float s = 0.f;
        #pragma unroll
        for (int d = 0; d < VQ; ++d) { hp[d] = row[d]; s += hp[d] * hp[d]; }
        float inv = 1.0f / (sqrtf(s) + 1e-8f);
        float hh  = s * inv * inv;              // ||hn||^2
        float dist[MCODES];
        #pragma unroll 1
        for (int m = 0; m < MCODES; ++m) {
            float dot = 0.f;
            #pragma unroll
            for (int d = 0; d < VQ; ++d) dot += hp[d] * EL[m * VQ + d];
            dist[m] = hh + E2[m] - 2.0f * inv * dot;
        }
        int g = tokBase + col;
        bool mk = (attn_mask[g] == 1);
        unsigned long long code = 0ull;
        float av[MCODES];
        #pragma unroll
        for (int p = 0; p < LOG2N; ++p) {
            float d0 = dist[2 * p], d1 = dist[2 * p + 1];
            float a1 = 1.0f / (1.0f + __expf(d1 - d0));   // softmax(-d) pairwise, stable
            float a0 = 1.0f - a1;
            if (d0 > d1) code |= (1ull << p);             // argmax, ties -> 0
            av[2 * p]     = mk ? a0 : 0.f;
            av[2 * p + 1] = mk ? a1 : 0.f;
        }
        outCode[g] = mk ? (long long)code : 0ll;
        // h_vq = maskedA @ E  (32x32, in place)
        #pragma unroll 1
        for (int d = 0; d < VQ; ++d) {
            float a = 0.f;
            for (int m = 0; m < MCODES; ++m) a += av[m] * EL[m * VQ + d];
            row[d] = a;
        }
    }

    // ---- GEMM2: out[16x512] = hvq[16x32] @ Wi[32x512] + bi, in 4 chunks of 128 cols ----
    for (int c = 0; c < 4; ++c) {
        __syncthreads();
        // swizzle-stage this chunk's Wi slice (16 KB)
        for (int f = tid; f < 4096; f += TPB) {
            int j = f & 1, ln = (f >> 1) & 31, nt = (f >> 6) & 7, kb = f >> 9;
            int kd = 4 * kb + 2 * (ln >> 4) + j;
            POOL[f] = Wi[kd * EMB + (c * 8 + nt) * 16 + (ln & 15)];
        }
        __syncthreads();
        v8f acc[8];
        #pragma unroll
        for (int nt = 0; nt < 8; ++nt) {
            float bv = bi[c * 128 + nt * 16 + col];
            #pragma unroll
            for (int r = 0; r < 8; ++r) acc[nt][r] = bv;
        }
        #pragma unroll 2
        for (int kb = 0; kb < 8; ++kb) {
            v2f a = *(const v2f*)&scrw[col * SCR_STRIDE + 4 * kb + 2 * kh];
            #pragma unroll
            for (int nt = 0; nt < 8; ++nt) {
                v2f b = *(const v2f*)&POOL[kb * 512 + nt * 64 + lane * 2];
                acc[nt] = __builtin_amdgcn_wmma_f32_16x16x4_f32(false, a, false, b, (short)0, acc[nt], false, false);
            }
        }
        #pragma unroll
        for (int nt = 0; nt < 8; ++nt) {
            #pragma unroll
            for (int r = 0; r < 8; ++r) {
                int tok = r + 8 * kh;
                outQ[(size_t)(tokBase + tok) * EMB + c * 128 + nt * 16 + col] = acc[nt][r];
            }
        }
    }

    if (blockIdx.x == 0 && tid == 0) *outLoss = 0.0f;
}

extern "C" void kernel_launch(void* const* d_in, const int* in_sizes, int n_in,
                              void* d_out, int out_size, void* d_ws, size_t ws_size,
                              hipStream_t stream) {
    (void)in_sizes; (void)n_in; (void)out_size; (void)d_ws; (void)ws_size;
    const float* h    = (const float*)d_in[0];
    const int*   mask = (const int*)  d_in[1];
    const float* Wp   = (const float*)d_in[2];
    const float* bp   = (const float*)d_in[3];
    const float* Wi   = (const float*)d_in[4];
    const float* bi   = (const float*)d_in[5];
    const float* emb  = (const float*)d_in[6];

    float* outQ = (float*)d_out;
    long long* outCode = (long long*)(outQ + (size_t)NTOK * EMB); // int64 section
    float* outLoss = (float*)(outCode + NTOK);                    // scalar loss

    dim3 grid(NTOK / TOK_PER_BLOCK);   // 512 blocks
    dim3 block(TPB);
    softgvq_fused<<<grid, block, 0, stream>>>(h, mask, Wp, bp, Wi, bi, emb,
                                              outQ, outCode, outLoss);
}